// InfluencerRankModel_42640435315019
// MI455X (gfx1250) — compile-verified
//
#include <hip/hip_runtime.h>
#include <hip/hip_bf16.h>

#define T_STEPS 8
#define N_NODES 40000
#define F_IN    256
#define E_EDGES 640000
#define G_DIM   128
#define B_TGT   1024

typedef __attribute__((ext_vector_type(2))) float v2f;
typedef __attribute__((ext_vector_type(8))) float v8f;

// ---------------------------------------------------------------------------
// C[M x 128] = relu(A[M x K] @ W[K x 128] + bias)   (row-major, M % 32 == 0)
// 256 threads = 8 wave32; block covers 32 rows; wave w covers cols [16w,16w+16)
// with TWO 16x16 M-tiles, so each B fragment feeds two V_WMMA_F32_16X16X4_F32.
// A-tile (32 x K, contiguous in memory) is filled with the CDNA5 async copy
// engine: global_load_async_to_lds_b128 + s_wait_asynccnt (ASYNCcnt-tracked).
// ---------------------------------------------------------------------------
__global__ __launch_bounds__(256) void gemm_bias_relu(
    const float* __restrict__ A, const float* __restrict__ W,
    const float* __restrict__ bias, float* __restrict__ C, int K)
{
    __shared__ float As[32 * 256];                 // 32 rows x K (K <= 256)
    const int tid  = threadIdx.x;
    const int row0 = blockIdx.x * 32;

    // rows are consecutive & full width -> the tile is one contiguous slab
    const float* gsrc = A + (size_t)row0 * K;
    const unsigned lds_base = (unsigned)(unsigned long long)As;  // LDS byte addr
    const int nchunk = 8 * K;                      // (32*K)/4 float4 chunks
    for (int c = tid; c < nchunk; c += 256) {
        asm volatile("global_load_async_to_lds_b128 %0, %1, off"
                     :: "v"(lds_base + (unsigned)c * 16u),
                        "v"(gsrc + (size_t)c * 4)
                     : "memory");
    }
    asm volatile("s_wait_asynccnt 0x0" ::: "memory");
    __syncthreads();

    const int wave = tid >> 5;
    const int lane = tid & 31;
    const int half = lane >> 4;                    // 0: lanes 0-15, 1: lanes 16-31
    const int l    = lane & 15;
    const int col  = wave * 16 + l;

    v8f acc0 = {0.f, 0.f, 0.f, 0.f, 0.f, 0.f, 0.f, 0.f};
    v8f acc1 = {0.f, 0.f, 0.f, 0.f, 0.f, 0.f, 0.f, 0.f};
    for (int k0 = 0; k0 < K; k0 += 4) {
        const int ka = k0 + 2 * half;              // A 16x4: lanes16-31 hold K+2,K+3
        v2f a0, a1, b;
        b[0]  = W[(size_t)ka * 128 + col];         // B 4x16: row striped across lanes
        b[1]  = W[(size_t)(ka + 1) * 128 + col];
        a0[0] = As[l * K + ka];
        a0[1] = As[l * K + ka + 1];
        a1[0] = As[(l + 16) * K + ka];
        a1[1] = As[(l + 16) * K + ka + 1];
        acc0 = __builtin_amdgcn_wmma_f32_16x16x4_f32(
                   false, a0, false, b, (short)0, acc0, false, false);
        acc1 = __builtin_amdgcn_wmma_f32_16x16x4_f32(
                   false, a1, false, b, (short)0, acc1, false, false);
    }

    const float bv = bias[col];
    #pragma unroll
    for (int r = 0; r < 8; ++r) {                  // C/D: VGPR r -> M = r + 8*half
        const int m0 = row0 + r + 8 * half;
        const int m1 = m0 + 16;
        float v0 = acc0[r] + bv;
        float v1 = acc1[r] + bv;
        C[(size_t)m0 * 128 + col] = v0 > 0.f ? v0 : 0.f;
        C[(size_t)m1 * 128 + col] = v1 > 0.f ? v1 : 0.f;
    }
}

// ---------------------------------------------------------------------------
// Degree / normalization helpers
// ---------------------------------------------------------------------------
__global__ void zero_f(float* __restrict__ p, int n)
{
    int i = blockIdx.x * 256 + threadIdx.x;
    if (i < n) p[i] = 0.f;
}

__global__ void deg_scatter(const int* __restrict__ dst, float* __restrict__ deg)
{
    int e = blockIdx.x * 256 + threadIdx.x;        // E is exact multiple of 256
    atomicAdd(&deg[dst[e]], 1.0f);
}

__global__ void make_dinv(float* __restrict__ d)   // deg -> rsqrt(deg + selfloop)
{
    int i = blockIdx.x * 256 + threadIdx.x;
    if (i < N_NODES) d[i] = rsqrtf(d[i] + 1.0f);
}

// agg[i,:] = h[i,:] * dinv[i]^2   (self-loop term), float4-vectorized
__global__ __launch_bounds__(256) void self_init(
    const float4* __restrict__ h, const float* __restrict__ dinv,
    float4* __restrict__ agg)
{
    int idx = blockIdx.x * 256 + threadIdx.x;      // over N*32 float4 chunks
    float di = dinv[idx >> 5];
    float w = di * di;
    float4 v = h[idx];
    agg[idx] = make_float4(v.x * w, v.y * w, v.z * w, v.w * w);
}

// agg[dst,:] += h[src,:] * dinv[src]*dinv[dst]  -- 32 lanes x float4 per edge
__global__ __launch_bounds__(256) void edge_scatter(
    const int* __restrict__ src, const int* __restrict__ dst,
    const float* __restrict__ dinv, const float* __restrict__ h,
    float* __restrict__ agg)
{
    const int e    = blockIdx.x * 8 + (threadIdx.x >> 5);
    const int lane = threadIdx.x & 31;
    const int s = src[e], d = dst[e];
    const float w = dinv[s] * dinv[d];
    const float4 hv = *(const float4*)(h + (size_t)s * 128 + lane * 4);
    float* ap = agg + (size_t)d * 128 + lane * 4;
    atomicAdd(ap + 0, hv.x * w);
    atomicAdd(ap + 1, hv.y * w);
    atomicAdd(ap + 2, hv.z * w);
    atomicAdd(ap + 3, hv.w * w);
}

// tgt[t][b][0:128] = h1[tg[b]];  tgt[t][b][128:256] = h2[tg[b]]
__global__ __launch_bounds__(256) void gather_targets(
    const float* __restrict__ h1, const float* __restrict__ h2,
    const int* __restrict__ tg, float* __restrict__ tgt, int t)
{
    const int b = blockIdx.x, j = threadIdx.x;
    const int node = tg[b];
    float v = (j < 128) ? h1[(size_t)node * 128 + j]
                        : h2[(size_t)node * 128 + (j - 128)];
    tgt[((size_t)t * B_TGT + b) * 256 + j] = v;
}

// ---------------------------------------------------------------------------
// Fused GRU (T=8) + attention + prediction MLP. One 128-thread block per target.
// Thread j owns hidden unit j; rnn_out kept in registers across all 8 steps.
// ---------------------------------------------------------------------------
__global__ __launch_bounds__(128) void gru_attn_pred(
    const float* __restrict__ tgt,                 // [T][B][256]
    const float* __restrict__ w_ih,                // [384][256]
    const float* __restrict__ w_hh,                // [384][128]
    const float* __restrict__ b_ih, const float* __restrict__ b_hh,
    const float* __restrict__ attn_w, const float* __restrict__ attn_b,
    const float* __restrict__ pw1, const float* __restrict__ pb1,
    const float* __restrict__ pw2, const float* __restrict__ pb2,
    float* __restrict__ out)                       // [B] pred ++ [B][T] attn
{
    __shared__ float xs[256];
    __shared__ float hs[128];
    __shared__ float red[128];
    __shared__ float sv[T_STEPS];
    __shared__ float hm[16];

    const int b = blockIdx.x;
    const int j = threadIdx.x;

    float hj = 0.f;
    float rr[T_STEPS];
    const float bir = b_ih[j], biz = b_ih[128 + j], big = b_ih[256 + j];
    const float bhr = b_hh[j], bhz = b_hh[128 + j], bhg = b_hh[256 + j];

    for (int t = 0; t < T_STEPS; ++t) {
        __syncthreads();
        hs[j] = hj;
        const float* xrow = tgt + ((size_t)t * B_TGT + b) * 256;
        xs[j]       = xrow[j];
        xs[j + 128] = xrow[j + 128];
        __syncthreads();

        float ir = bir, iz = biz, ig = big;
        const float* wr = w_ih + (size_t)j * 256;
        const float* wz = w_ih + (size_t)(128 + j) * 256;
        const float* wg = w_ih + (size_t)(256 + j) * 256;
        for (int k = 0; k < 256; ++k) {
            const float xv = xs[k];
            ir += xv * wr[k]; iz += xv * wz[k]; ig += xv * wg[k];
        }
        float hr = bhr, hz = bhz, hg = bhg;
        const float* vr = w_hh + (size_t)j * 128;
        const float* vz = w_hh + (size_t)(128 + j) * 128;
        const float* vg = w_hh + (size_t)(256 + j) * 128;
        for (int k = 0; k < 128; ++k) {
            const float hv = hs[k];
            hr += hv * vr[k]; hz += hv * vz[k]; hg += hv * vg[k];
        }
        const float r = 1.f / (1.f + expf(-(ir + hr)));
        const float z = 1.f / (1.f + expf(-(iz + hz)));
        const float g = tanhf(ig + r * hg);
        hj = (1.f - z) * g + z * hj;
        rr[t] = hj;
    }

    // attention scores: s_t = tanh(<rnn_t, attn_w> + attn_b)
    const float aw = attn_w[j];
    const float ab = attn_b[0];
    for (int t = 0; t < T_STEPS; ++t) {
        __syncthreads();
        red[j] = rr[t] * aw;
        __syncthreads();
        for (int s = 64; s > 0; s >>= 1) {
            if (j < s) red[j] += red[j + s];
            __syncthreads();
        }
        if (j == 0) sv[t] = tanhf(red[0] + ab);
    }
    __syncthreads();

    float smax = sv[0];
    for (int t = 1; t < T_STEPS; ++t) smax = fmaxf(smax, sv[t]);
    float ev[T_STEPS], esum = 0.f;
    for (int t = 0; t < T_STEPS; ++t) { ev[t] = expf(sv[t] - smax); esum += ev[t]; }
    const float inv = 1.f / esum;

    float rep = 0.f;
    for (int t = 0; t < T_STEPS; ++t) rep += rr[t] * (ev[t] * inv);
    if (j < T_STEPS) out[B_TGT + b * T_STEPS + j] = ev[j] * inv;

    __syncthreads();
    red[j] = rep;                                  // rep vector in LDS
    __syncthreads();
    if (j < 16) {
        float acc = pb1[j];
        for (int k = 0; k < 128; ++k) acc += red[k] * pw1[k * 16 + j];
        hm[j] = acc > 0.f ? acc : 0.f;
    }
    __syncthreads();
    if (j == 0) {
        float p = pb2[0];
        for (int i = 0; i < 16; ++i) p += hm[i] * pw2[i];
        out[b] = p;
    }
}

// ---------------------------------------------------------------------------
extern "C" void kernel_launch(void* const* d_in, const int* in_sizes, int n_in,
                              void* d_out, int out_size, void* d_ws, size_t ws_size,
                              hipStream_t stream)
{
    (void)in_sizes; (void)n_in; (void)out_size; (void)ws_size;

    const float* x      = (const float*)d_in[0];
    const int*   ei     = (const int*)d_in[1];
    const int*   tg     = (const int*)d_in[2];
    const float* proj_w = (const float*)d_in[3];
    const float* proj_b = (const float*)d_in[4];
    const float* gw1    = (const float*)d_in[5];
    const float* gb1    = (const float*)d_in[6];
    const float* gw2    = (const float*)d_in[7];
    const float* gb2    = (const float*)d_in[8];
    const float* wih    = (const float*)d_in[9];
    const float* whh    = (const float*)d_in[10];
    const float* bih    = (const float*)d_in[11];
    const float* bhh    = (const float*)d_in[12];
    const float* aw     = (const float*)d_in[13];
    const float* ab     = (const float*)d_in[14];
    const float* pw1    = (const float*)d_in[15];
    const float* pb1    = (const float*)d_in[16];
    const float* pw2    = (const float*)d_in[17];
    const float* pb2    = (const float*)d_in[18];
    float* out = (float*)d_out;

    // workspace layout (floats): dinv | bufA | bufB | bufC | tgt  (~70 MB)
    float* ws = (float*)d_ws;
    const size_t NH = (size_t)N_NODES * 128;
    float* dinv = ws;
    float* bufA = ws + 40960;          // h (proj out), later h2
    float* bufB = bufA + NH;           // agg (both layers)
    float* bufC = bufB + NH;           // h1
    float* tgt  = bufC + NH;           // [T][B][256]

    const int* e_src = ei;             // edge_index[0]
    const int* e_dst = ei + E_EDGES;   // edge_index[1]

    // normalization: dinv = rsqrt(in-degree + self-loop)
    zero_f<<<(N_NODES + 255) / 256, 256, 0, stream>>>(dinv, N_NODES);
    deg_scatter<<<E_EDGES / 256, 256, 0, stream>>>(e_dst, dinv);
    make_dinv<<<(N_NODES + 255) / 256, 256, 0, stream>>>(dinv);

    const int nq = (int)(NH / 4);      // float4 chunks of an N x 128 buffer
    for (int t = 0; t < T_STEPS; ++t) {
        const float* xt = x + (size_t)t * N_NODES * F_IN;
        // h = relu(x_t @ proj_w + proj_b)
        gemm_bias_relu<<<N_NODES / 32, 256, 0, stream>>>(xt, proj_w, proj_b, bufA, F_IN);
        // agg1 = D^-1/2 (A+I) D^-1/2 h   (aggregate-then-matmul: linearity)
        self_init<<<nq / 256, 256, 0, stream>>>((const float4*)bufA, dinv, (float4*)bufB);
        edge_scatter<<<E_EDGES / 8, 256, 0, stream>>>(e_src, e_dst, dinv, bufA, bufB);
        // h1 = relu(agg1 @ w1 + b1)
        gemm_bias_relu<<<N_NODES / 32, 256, 0, stream>>>(bufB, gw1, gb1, bufC, G_DIM);
        // agg2 from h1
        self_init<<<nq / 256, 256, 0, stream>>>((const float4*)bufC, dinv, (float4*)bufB);
        edge_scatter<<<E_EDGES / 8, 256, 0, stream>>>(e_src, e_dst, dinv, bufC, bufB);
        // h2 = relu(agg2 @ w2 + b2)
        gemm_bias_relu<<<N_NODES / 32, 256, 0, stream>>>(bufB, gw2, gb2, bufA, G_DIM);
        // tgt[t] = concat(h1, h2)[target_indices]
        gather_targets<<<B_TGT, 256, 0, stream>>>(bufC, bufA, tg, tgt, t);
    }

    gru_attn_pred<<<B_TGT, 128, 0, stream>>>(tgt, wih, whh, bih, bhh,
                                             aw, ab, pw1, pb1, pw2, pb2, out);
}